// Symmetrizer_JIT_2843268350082
// MI455X (gfx1250) — compile-verified
//
#include <hip/hip_runtime.h>
#include <hip/hip_bf16.h>

// ---------------------------------------------------------------------------
// Compile-time regeneration of the Symmetrizer tables for MAX_L = 3
// (exact mirror of the Python _build_tables logic), plus:
//   1) symmetric dedup: identical sorted monomials in a segment merged with
//      summed prefactors (108 -> 70 triple terms)
//   2) Horner grouping on the smallest index: seg-wise factoring
//      sum p*x[i0]*x[i1]*x[i2] = sum_{i0} x[i0] * (sum p*x[i1]*x[i2])
//      (70 terms -> 18 groups; kills the per-term second multiply)
// ---------------------------------------------------------------------------
namespace sym {

constexpr int MAXL = 3;
constexpr int NANG = 20;   // 1 + 3 + 6 + 10
constexpr int N2   = 19;   // 3 + 6 + 10
constexpr int N3   = 108;  // raw, pre-dedup
constexpr int NG2  = 3;
constexpr int NG3  = 4;

constexpr int fact(int n) { return n <= 1 ? 1 : n * fact(n - 1); }
constexpr int multinom(int a, int b, int c) {
    return fact(a + b + c) / (fact(a) * fact(b) * fact(c));
}

constexpr int lidx(int x, int y, int z) {
    int i = 0;
    for (int L = 0; L <= MAXL; ++L)
        for (int lx = L; lx >= 0; --lx)
            for (int ly = L - lx; ly >= 0; --ly) {
                if (lx == x && ly == y && (L - lx - ly) == z) return i;
                ++i;
            }
    return -1;
}

struct Tab2 { int idx[N2]; int seg[N2]; float pref[N2]; };

constexpr Tab2 make2() {
    Tab2 t{};
    int e = 0;
    for (int L = 1; L <= MAXL; ++L)
        for (int a = 0; a <= L; ++a)
            for (int b = 0; b <= L - a; ++b) {
                const int c = L - a - b;
                t.idx[e]  = lidx(a, b, c);
                t.seg[e]  = L - 1;
                t.pref[e] = (float)multinom(a, b, c);
                ++e;
            }
    return t;
}

constexpr void csort3(int& a, int& b, int& c) {
    if (a > b) { int t = a; a = b; b = t; }
    if (b > c) { int t = b; b = c; c = t; }
    if (a > b) { int t = a; a = b; b = t; }
}

struct Tab3 {
    int i0[N3]; int i1[N3]; int i2[N3]; int seg[N3]; float pref[N3];
    int n;
};

constexpr int count3_raw() {
    int e = 0;
    for (int a = 1; a <= MAXL; ++a)
        for (int b = 0; b <= a; ++b)
            for (int c = 0; c <= b; ++c)
                if (b + c >= 1 && a + b <= MAXL && b + c <= MAXL)
                    e += ((a + 1) * (a + 2) / 2) * ((b + 1) * (b + 2) / 2) *
                         ((c + 1) * (c + 2) / 2);
    return e;
}
static_assert(count3_raw() == N3, "raw triple-term count mismatch");

constexpr Tab3 make3_dedup() {
    Tab3 t{};
    t.n = 0;
    int g = 0;
    for (int a = 1; a <= MAXL; ++a)
        for (int b = 0; b <= a; ++b)
            for (int c = 0; c <= b; ++c) {
                if (!(b + c >= 1 && a + b <= MAXL && b + c <= MAXL)) continue;
                for (int ax = 0; ax <= a; ++ax)
                for (int ay = 0; ay <= a - ax; ++ay) {
                    const int az = a - ax - ay;
                    for (int bx = 0; bx <= b; ++bx)
                    for (int by = 0; by <= b - bx; ++by) {
                        const int bz = b - bx - by;
                        for (int cx = 0; cx <= c; ++cx)
                        for (int cy = 0; cy <= c - cx; ++cy) {
                            const int cz = c - cx - cy;
                            int v1 = lidx(ax + bx, ay + by, az + bz);
                            int v2 = lidx(ax + cx, ay + cy, az + cz);
                            int v3 = lidx(bx + cx, by + cy, bz + cz);
                            csort3(v1, v2, v3);
                            const float p =
                                (float)(multinom(ax, ay, az) *
                                        multinom(bx, by, bz) *
                                        multinom(cx, cy, cz));
                            int found = -1;
                            for (int k = 0; k < t.n; ++k)
                                if (t.seg[k] == g && t.i0[k] == v1 &&
                                    t.i1[k] == v2 && t.i2[k] == v3) {
                                    found = k; break;
                                }
                            if (found >= 0) {
                                t.pref[found] += p;
                            } else {
                                t.i0[t.n] = v1; t.i1[t.n] = v2; t.i2[t.n] = v3;
                                t.seg[t.n] = g; t.pref[t.n] = p;
                                ++t.n;
                            }
                        }
                    }
                }
                ++g;
            }
    return t;
}

constexpr Tab2 T2  = make2();
constexpr Tab3 T3  = make3_dedup();
constexpr int  N3D = T3.n;
static_assert(N3D == 70, "dedup count changed -- re-derive expectation");

// --- Horner grouping on (seg, i0) ------------------------------------------
struct GTab3 {
    int   gi0[N3]; int gseg[N3]; int start[N3 + 1]; int ngrp;
    int   j1[N3];  int j2[N3];   float p[N3];
};

constexpr GTab3 make_g3() {
    GTab3 g{};
    g.ngrp = 0;
    int pos = 0;
    for (int s = 0; s < NG3; ++s)
        for (int i0v = 0; i0v < NANG; ++i0v) {
            bool any = false;
            for (int k = 0; k < T3.n; ++k)
                if (T3.seg[k] == s && T3.i0[k] == i0v) {
                    if (!any) {
                        g.gi0[g.ngrp]  = i0v;
                        g.gseg[g.ngrp] = s;
                        g.start[g.ngrp] = pos;
                        any = true;
                    }
                    g.j1[pos] = T3.i1[k];
                    g.j2[pos] = T3.i2[k];
                    g.p[pos]  = T3.pref[k];
                    ++pos;
                }
            if (any) ++g.ngrp;
        }
    g.start[g.ngrp] = pos;
    return g;
}

constexpr GTab3 G3 = make_g3();
static_assert(G3.ngrp == 18, "group count changed -- re-derive expectation");
static_assert(G3.start[G3.ngrp] == N3D, "grouping lost terms");

} // namespace sym

// ---------------------------------------------------------------------------
// Kernel: one thread = one (node*radial row, channel-pair).
// 20 coalesced b64 loads, fully unrolled <2 x float> fma body (packed /
// dual-issue fp32 on gfx1250), 8 coalesced b64 stores.
// ---------------------------------------------------------------------------
typedef float v2f __attribute__((ext_vector_type(2)));

constexpr int N_NODES   = 10000;
constexpr int N_RADIAL  = 8;
constexpr int N_CHANNEL = 32;
constexpr int C2        = N_CHANNEL / 2;                    // 16 float2 per row
constexpr int NTOT      = N_NODES * N_RADIAL * C2;          // 1,280,000 threads
constexpr int NGOUT     = 1 + sym::NG2 + sym::NG3;          // 8 output groups

__global__ __launch_bounds__(256)
void symmetrizer_kernel(const v2f* __restrict__ in, v2f* __restrict__ out) {
    const int t = blockIdx.x * blockDim.x + threadIdx.x;
    if (t >= NTOT) return;
    const int c2 = t & (C2 - 1);
    const int nr = t >> 4;                       // node*8 + radial, 0..79999

    const v2f* __restrict__ src = in + (size_t)nr * (sym::NANG * C2) + c2;
    v2f x[sym::NANG];
#pragma unroll
    for (int a = 0; a < sym::NANG; ++a) x[a] = src[(size_t)a * C2];

    v2f acc[sym::NG2 + sym::NG3];
#pragma unroll
    for (int g = 0; g < sym::NG2 + sym::NG3; ++g) acc[g] = (v2f)0.0f;

    // b2: weighted sums of squares (x*x expressions also CSE with b3 pairs)
#pragma unroll
    for (int e = 0; e < sym::N2; ++e) {
        const v2f xa = x[sym::T2.idx[e]];
        acc[sym::T2.seg[e]] =
            __builtin_elementwise_fma((v2f)sym::T2.pref[e], xa * xa,
                                      acc[sym::T2.seg[e]]);
    }

    // b3: Horner-grouped monomials:
    //   acc[seg] += x[i0] * (sum_e p_e * x[j1_e]*x[j2_e])
#pragma unroll
    for (int gi = 0; gi < sym::G3.ngrp; ++gi) {
        const int st = sym::G3.start[gi];
        const int en = sym::G3.start[gi + 1];
        // first term: p*(pair); mul by 1.0 folds away exactly
        v2f s = (v2f)sym::G3.p[st] * (x[sym::G3.j1[st]] * x[sym::G3.j2[st]]);
#pragma unroll
        for (int e = st + 1; e < en; ++e)
            s = __builtin_elementwise_fma((v2f)sym::G3.p[e],
                                          x[sym::G3.j1[e]] * x[sym::G3.j2[e]],
                                          s);
        const int g = sym::NG2 + sym::G3.gseg[gi];
        acc[g] = __builtin_elementwise_fma(x[sym::G3.gi0[gi]], s, acc[g]);
    }

    v2f* __restrict__ dst = out + (size_t)nr * (NGOUT * C2) + c2;
    dst[0] = x[0];                               // b1
#pragma unroll
    for (int g = 0; g < sym::NG2 + sym::NG3; ++g)
        dst[(size_t)(g + 1) * C2] = acc[g];
}

// ---------------------------------------------------------------------------
extern "C" void kernel_launch(void* const* d_in, const int* in_sizes, int n_in,
                              void* d_out, int out_size, void* d_ws, size_t ws_size,
                              hipStream_t stream) {
    (void)in_sizes; (void)n_in; (void)out_size; (void)d_ws; (void)ws_size;
    const v2f* in  = (const v2f*)d_in[0];  // node_attr (tables are hardcoded constexpr)
    v2f*       out = (v2f*)d_out;
    const int blocks = (NTOT + 255) / 256; // 5000, exact
    symmetrizer_kernel<<<blocks, 256, 0, stream>>>(in, out);
}